// MultiHeadAttentionQuantum_65481071407544
// MI455X (gfx1250) — compile-verified
//
#include <hip/hip_runtime.h>

// ---- problem constants (from reference) ----
#define NUM_HEADS 16
#define D_K 8
#define EMBED 128          // NUM_HEADS * D_K
#define BATCH 8
#define SEQL 1024

typedef float v2f __attribute__((ext_vector_type(2)));
typedef float v8f __attribute__((ext_vector_type(8)));

// D(16x16 f32) = A(16x4 f32) * B(4x16 f32) + C
__device__ __forceinline__ v8f wmma_f32_k4(v2f a, v2f b, v8f c) {
  return __builtin_amdgcn_wmma_f32_16x16x4_f32(
      /*neg_a=*/false, a, /*neg_b=*/false, b,
      /*c_mod=*/(short)0, c, /*reuse_a=*/false, /*reuse_b=*/false);
}

// ---------------------------------------------------------------------------
// Kernel 1: quantum measurement z = f(x + theta), written head-major [b,h,s,8]
// z[0] = cos(a1)*...*cos(a7); z[w>=1] = cos(a0)*...*cos(aw)
// ---------------------------------------------------------------------------
__global__ __launch_bounds__(256) void qheads_kernel(
    const float* __restrict__ x, const float* __restrict__ theta,
    float* __restrict__ z) {
  int t = blockIdx.x * 256 + threadIdx.x;            // one 8-wide head vector
  if (t >= BATCH * NUM_HEADS * SEQL) return;
  int s = t & (SEQL - 1);
  int h = (t >> 10) & (NUM_HEADS - 1);
  int b = t >> 14;
  const float* xp = x + ((size_t)(b * SEQL + s)) * EMBED + h * D_K;
  float c[8];
#pragma unroll
  for (int i = 0; i < 8; ++i) c[i] = __cosf(xp[i] + theta[i]);
  float zv[8];
  float p17 = c[1];
#pragma unroll
  for (int i = 2; i < 8; ++i) p17 *= c[i];
  zv[0] = p17;                                       // wire 0
  float cp = c[0];
#pragma unroll
  for (int w = 1; w < 8; ++w) { cp *= c[w]; zv[w] = cp; }
  float* zp = z + (size_t)t * 8;
#pragma unroll
  for (int i = 0; i < 8; ++i) zp[i] = zv[i];
}

// ---------------------------------------------------------------------------
// Kernel 2: self-attention per (b,h), q=k=v=z[bh] (1024x8).
// Scores are bounded (|s| <= 8/sqrt(8) = 2.83) so no running max is needed:
// accumulate O = sum exp(s)*v and l = sum exp(s), normalize at the end.
// z slab staged into LDS row-padded to 16 floats (cols 8..15 = 0) via async
// global->LDS b128 copies, so all WMMA B-operand loads are unconditional.
// Scores: 2 chained f32 WMMAs (K=4); P restaged D->A layout via per-wave LDS;
// PV: 4 f32 WMMAs in two independent accumulator chains.
// ---------------------------------------------------------------------------
__global__ __launch_bounds__(256) void attn_kernel(
    const float* __restrict__ z, float* __restrict__ o) {
  __shared__ float zl[SEQL * 16];        // 64 KB: padded slab, cols 8..15 = 0
  __shared__ float pl[8][16][20];        // per-wave P staging (padded rows)

  int bh = blockIdx.x;                   // 0..127
  int b  = bh >> 4;
  int h  = bh & 15;
  const float* zp = z + (size_t)bh * SEQL * D_K;

  // zero-fill the pad halves (bytes 32..63 of each 64B row)
  for (int r = threadIdx.x; r < SEQL; r += 256) {
    float4 z4 = make_float4(0.f, 0.f, 0.f, 0.f);
    *(float4*)(zl + r * 16 + 8)  = z4;
    *(float4*)(zl + r * 16 + 12) = z4;
  }
  // async-stage the data halves: row i global [i*32, i*32+32) -> LDS [i*64, +32)
  {
    uint32_t zl_base =
        (uint32_t)(uintptr_t)(__attribute__((address_space(3))) float*)zl;
    for (int i = threadIdx.x; i < SEQL; i += 256) {
      unsigned long long ga = (unsigned long long)(uintptr_t)(zp + i * 8);
      uint32_t la = zl_base + i * 64;
      asm volatile("global_load_async_to_lds_b128 %0, %1, off"
                   :: "v"(la), "v"(ga) : "memory");
      asm volatile("global_load_async_to_lds_b128 %0, %1, off offset:16"
                   :: "v"(la), "v"(ga) : "memory");
    }
    asm volatile("s_wait_asynccnt 0" ::: "memory");
  }
  __syncthreads();

  const int lane = threadIdx.x & 31;
  const int wave = threadIdx.x >> 5;
  const int lrow = lane & 15;            // M (or N) index for this lane
  const int lh   = lane >> 4;            // lane half
  const int kb   = lh * 2;               // K-pair base within a K=4 chunk
  const float c_exp = 0.51006972f;       // log2(e) / sqrt(8)

  for (int qt = wave; qt < SEQL / 16; qt += 8) {
    // A operands for scores: Q rows (16x8), split into two K=4 chunks
    const float* Q = zl + (qt * 16 + lrow) * 16;
    v2f a0 = { Q[kb],     Q[kb + 1] };
    v2f a1 = { Q[kb + 4], Q[kb + 5] };

    v8f accA = {}, accB = {};            // two independent O chains
    float psum[8];                       // per-lane share of the row sums
#pragma unroll
    for (int r = 0; r < 8; ++r) psum[r] = 0.0f;

    for (int kt = 0; kt < SEQL / 16; ++kt) {
      // B operands: B[d][n] = K[n][d] -> same addressing pattern as A
      const float* Kt = zl + (kt * 16 + lrow) * 16;
      v2f b0 = { Kt[kb],     Kt[kb + 1] };
      v2f b1 = { Kt[kb + 4], Kt[kb + 5] };
      v8f sc = {};
      sc = wmma_f32_k4(a0, b0, sc);
      sc = wmma_f32_k4(a1, b1, sc);

      // P = exp2(s * log2e/sqrt(d)); bounded, no max subtraction needed
#pragma unroll
      for (int r = 0; r < 8; ++r) {
        float p = __builtin_amdgcn_exp2f(sc[r] * c_exp);
        sc[r] = p;
        psum[r] += p;
      }

      // re-layout P: D layout -> LDS -> A layout (per-wave region, in-order DS)
#pragma unroll
      for (int r = 0; r < 8; ++r) pl[wave][r + lh * 8][lrow] = sc[r];
      __builtin_amdgcn_wave_barrier();

      // O += P(16x16) * V(16x16-padded): 4 K=4 WMMAs, two accumulator chains
#pragma unroll
      for (int cc = 0; cc < 4; ++cc) {
        int kk = cc * 4 + kb;
        v2f pa = { pl[wave][lrow][kk], pl[wave][lrow][kk + 1] };
        const float* Vp = zl + (kt * 16 + kk) * 16 + lrow;   // rows kk, kk+1
        v2f vb = { Vp[0], Vp[16] };      // lanes 8..15 read the zero pad
        if (cc & 1) accB = wmma_f32_k4(pa, vb, accB);
        else        accA = wmma_f32_k4(pa, vb, accA);
      }
      __builtin_amdgcn_wave_barrier();   // keep next stores after these loads
    }

    // finish row sums: one 16-lane butterfly per q-tile
#pragma unroll
    for (int r = 0; r < 8; ++r) {
      float s = psum[r];
      s += __shfl_xor(s, 1, 32);
      s += __shfl_xor(s, 2, 32);
      s += __shfl_xor(s, 4, 32);
      s += __shfl_xor(s, 8, 32);
      psum[r] = s;
    }

    // epilogue: normalize and scatter to [b, s, e]
    if (lrow < D_K) {
#pragma unroll
      for (int r = 0; r < 8; ++r) {
        int row = qt * 16 + r + lh * 8;
        o[((size_t)(b * SEQL + row)) * EMBED + h * D_K + lrow] =
            (accA[r] + accB[r]) / psum[r];
      }
    }
  }
}

// ---------------------------------------------------------------------------
// Kernel 3: combine GEMM  Y[8192,128] = A[8192,128] @ W[128,128]^T
// Block = one 16-row M tile; wave w = N tile w. 32 chained K=4 f32 WMMAs.
// ---------------------------------------------------------------------------
__global__ __launch_bounds__(256) void combine_kernel(
    const float* __restrict__ A, const float* __restrict__ W,
    float* __restrict__ Y) {
  int mt   = blockIdx.x;                 // 0..511
  int wave = threadIdx.x >> 5;           // 0..7 -> N tile
  int lane = threadIdx.x & 31;
  int lrow = lane & 15;
  int lh   = lane >> 4;
  int m0 = mt * 16, n0 = wave * 16;
  const float* Ap = A + (size_t)(m0 + lrow) * EMBED;   // A rows
  const float* Wp = W + (size_t)(n0 + lrow) * EMBED;   // B[k][n] = W[n][k]
  v8f acc = {};
#pragma unroll 8
  for (int kc = 0; kc < EMBED / 4; ++kc) {
    int k = kc * 4 + lh * 2;
    v2f a = { Ap[k], Ap[k + 1] };
    v2f bb = { Wp[k], Wp[k + 1] };
    acc = wmma_f32_k4(a, bb, acc);
  }
#pragma unroll
  for (int r = 0; r < 8; ++r)
    Y[(size_t)(m0 + r + lh * 8) * EMBED + n0 + lrow] = acc[r];
}

// ---------------------------------------------------------------------------
extern "C" void kernel_launch(void* const* d_in, const int* in_sizes, int n_in,
                              void* d_out, int out_size, void* d_ws, size_t ws_size,
                              hipStream_t stream) {
  const float* x     = (const float*)d_in[0];   // [8,1024,128]
  const float* theta = (const float*)d_in[1];   // [8]
  const float* wc    = (const float*)d_in[2];   // [128,128]
  float* out = (float*)d_out;                   // [8,1024,128]

  float* z    = (float*)d_ws;                                   // 4 MB: [b,h,s,8]
  float* attn = z + (size_t)BATCH * NUM_HEADS * SEQL * D_K;     // 4 MB: [b,s,e]

  // 1) quantum heads: 131072 vectors, one per thread
  qheads_kernel<<<(BATCH * NUM_HEADS * SEQL) / 256, 256, 0, stream>>>(x, theta, z);

  // 2) attention per (b,h)
  attn_kernel<<<BATCH * NUM_HEADS, 256, 0, stream>>>(z, attn);

  // 3) output projection
  combine_kernel<<<(BATCH * SEQL) / 16, 256, 0, stream>>>(attn, wc, out);
}